// SynapticCausalSelfAttention_3229815406730
// MI455X (gfx1250) — compile-verified
//
#include <hip/hip_runtime.h>
#include <hip/hip_bf16.h>

// ---------------------------------------------------------------------------
// SynapticCausalSelfAttention for MI455X (gfx1250), fp32 WMMA throughout,
// async global->LDS tile staging (GLOBAL_LOAD_ASYNC_TO_LDS_B128).
// B=2, T=1024, H=16, D=64, NE=1024
// ---------------------------------------------------------------------------

#define Bc 2
#define Tc 1024
#define Hc 16
#define Dc 64
#define NEc 1024
#define EPSf 1e-6f

typedef __attribute__((ext_vector_type(2))) float v2f;
typedef __attribute__((ext_vector_type(8))) float v8f;
typedef int v4i __attribute__((vector_size(16)));   // matches builtin param type

#ifndef __has_builtin
#define __has_builtin(x) 0
#endif
#if __has_builtin(__builtin_amdgcn_global_load_async_to_lds_b128) && \
    __has_builtin(__builtin_amdgcn_s_wait_asynccnt)
#define USE_ASYNC_LDS 1
#else
#define USE_ASYNC_LDS 0
#endif

#define AS1 __attribute__((address_space(1)))
#define AS3 __attribute__((address_space(3)))

// Copy 4 consecutive floats global -> LDS (async DMA if available).
__device__ __forceinline__ void cp_f4(float* dst_lds, const float* src_g) {
#if USE_ASYNC_LDS
  __builtin_amdgcn_global_load_async_to_lds_b128((AS1 v4i*)src_g,
                                                 (AS3 v4i*)dst_lds, 0, 0);
#else
  float4 t = *(const float4*)src_g;
  *(float4*)dst_lds = t;
#endif
}

__device__ __forceinline__ void cp_wait() {
#if USE_ASYNC_LDS
  __builtin_amdgcn_s_wait_asynccnt(0);
#endif
}

// D = A(16x4) * B(4x16) + C, all fp32 (V_WMMA_F32_16X16X4_F32).
__device__ __forceinline__ v8f wmma_f32(v2f a, v2f b, v8f c) {
  return __builtin_amdgcn_wmma_f32_16x16x4_f32(false, a, false, b,
                                               (short)0, c, false, false);
}

__device__ __forceinline__ v8f v8f_zero() {
  v8f z = {0.f, 0.f, 0.f, 0.f, 0.f, 0.f, 0.f, 0.f};
  return z;
}

__device__ __forceinline__ float warp_sum(float v) {
  #pragma unroll
  for (int o = 16; o > 0; o >>= 1) v += __shfl_xor(v, o, 32);
  return v;
}

__device__ __forceinline__ float sigmoidf_(float x) {
  return 1.f / (1.f + expf(-x));
}

// ---------------------------------------------------------------------------
// Kernel 1: fused  q = rmsnorm(rope(x@Wq)), k = rmsnorm(rope(x@Wk)), v = x@Wv
// grid (B*T/16, H), block 128 (4 waves). Output layout (B,H,T,D).
// ---------------------------------------------------------------------------
__global__ __launch_bounds__(128)
void qkv_kernel(const float* __restrict__ x,
                const float* __restrict__ Wq,
                const float* __restrict__ Wk,
                const float* __restrict__ Wv,
                const float* __restrict__ rc,
                const float* __restrict__ rs,
                float* __restrict__ q, float* __restrict__ k,
                float* __restrict__ v) {
  __shared__ float xs[16][16];
  __shared__ float wsq[16][64], wsk[16][64], wsv[16][64];
  __shared__ float qA[16][64], kA[16][64];
  __shared__ float rinvq[16], rinvk[16];

  const int tid  = threadIdx.x;
  const int wave = tid >> 5;
  const int lane = tid & 31;
  const int row0 = blockIdx.x * 16;          // flat (b*T + t) row base
  const int h    = blockIdx.y;
  const int n0   = wave * 16;
  const int ml   = lane & 15;
  const int kb   = (lane >> 4) * 2;

  v8f aq = v8f_zero(), ak = v8f_zero(), av = v8f_zero();

  for (int k0 = 0; k0 < NEc; k0 += 16) {
    // x tile 16x16: 64 f4-chunks
    for (int e = tid; e < 64; e += 128) {
      int r = e >> 2, c = (e & 3) * 4;
      cp_f4(&xs[r][c], &x[(size_t)(row0 + r) * NEc + k0 + c]);
    }
    // W tiles 16x64 each: 256 f4-chunks per matrix
    for (int e = tid; e < 256; e += 128) {
      int r = e >> 4, c = (e & 15) * 4;
      size_t gi = (size_t)(k0 + r) * NEc + h * Dc + c;
      cp_f4(&wsq[r][c], &Wq[gi]);
      cp_f4(&wsk[r][c], &Wk[gi]);
      cp_f4(&wsv[r][c], &Wv[gi]);
    }
    cp_wait();
    __syncthreads();
    #pragma unroll
    for (int kk = 0; kk < 16; kk += 4) {
      v2f a;  a.x  = xs[ml][kk + kb];            a.y  = xs[ml][kk + kb + 1];
      v2f bq; bq.x = wsq[kk + kb][n0 + ml];      bq.y = wsq[kk + kb + 1][n0 + ml];
      v2f bk; bk.x = wsk[kk + kb][n0 + ml];      bk.y = wsk[kk + kb + 1][n0 + ml];
      v2f bv; bv.x = wsv[kk + kb][n0 + ml];      bv.y = wsv[kk + kb + 1][n0 + ml];
      aq = wmma_f32(a, bq, aq);
      ak = wmma_f32(a, bk, ak);
      av = wmma_f32(a, bv, av);
    }
    __syncthreads();
  }

  const int b  = row0 / Tc;
  const int t0 = row0 % Tc;
  const int hi = lane >> 4;
  #pragma unroll
  for (int r = 0; r < 8; r++) {
    int mrow = r + hi * 8;
    qA[mrow][n0 + ml] = aq[r];
    kA[mrow][n0 + ml] = ak[r];
    v[(((size_t)b * Hc + h) * Tc + t0 + mrow) * Dc + n0 + ml] = av[r];
  }
  __syncthreads();

  // RoPE in place (each task writes the pair d, d+32 it reads)
  for (int e = tid; e < 512; e += 128) {
    int m = e >> 5, d = e & 31;
    int t = t0 + m;
    float c = rc[t * 32 + d], s = rs[t * 32 + d];
    float q1 = qA[m][d], q2 = qA[m][d + 32];
    qA[m][d]      = q1 * c - q2 * s;
    qA[m][d + 32] = q1 * s + q2 * c;
    float k1 = kA[m][d], k2 = kA[m][d + 32];
    kA[m][d]      = k1 * c - k2 * s;
    kA[m][d + 32] = k1 * s + k2 * c;
  }
  __syncthreads();

  if (tid < 16) {
    float sq = 0.f, sk = 0.f;
    #pragma unroll
    for (int d = 0; d < 64; d++) {
      sq += qA[tid][d] * qA[tid][d];
      sk += kA[tid][d] * kA[tid][d];
    }
    rinvq[tid] = rsqrtf(sq * (1.f / 64.f) + EPSf);
    rinvk[tid] = rsqrtf(sk * (1.f / 64.f) + EPSf);
  }
  __syncthreads();

  for (int e = tid; e < 1024; e += 128) {
    int m = e >> 6, d = e & 63;
    size_t o = (((size_t)b * Hc + h) * Tc + t0 + m) * Dc + d;
    q[o] = qA[m][d] * rinvq[m];
    k[o] = kA[m][d] * rinvk[m];
  }
}

// ---------------------------------------------------------------------------
// Kernel 2: S = q k^T / sqrt(D), writes both S and S^T (full T x T per head).
// grid (B*H, T/64, T/64), block 256 (8 waves, 2 sub-tiles each).
// ---------------------------------------------------------------------------
__global__ __launch_bounds__(256)
void scores_kernel(const float* __restrict__ q, const float* __restrict__ k,
                   float* __restrict__ S, float* __restrict__ ST) {
  __shared__ float qs[64][64];
  __shared__ float ks[64][64];

  const int bh = blockIdx.x;
  const int i0 = blockIdx.y * 64;
  const int j0 = blockIdx.z * 64;
  const int tid = threadIdx.x, lane = tid & 31, wave = tid >> 5;
  const int ml = lane & 15, kb2 = (lane >> 4) * 2, hi = lane >> 4;

  const float* qb = q + (size_t)bh * Tc * Dc;
  const float* kb_ = k + (size_t)bh * Tc * Dc;
  for (int e = tid; e < 1024; e += 256) {     // 64x64 floats = 1024 f4-chunks
    int r = e >> 4, c = (e & 15) * 4;
    cp_f4(&qs[r][c], &qb[(size_t)(i0 + r) * Dc + c]);
    cp_f4(&ks[r][c], &kb_[(size_t)(j0 + r) * Dc + c]);
  }
  cp_wait();
  __syncthreads();

  const size_t base = (size_t)bh * Tc * Tc;
  for (int tt = wave; tt < 16; tt += 8) {
    int ti = tt >> 2, tj = tt & 3;
    v8f acc = v8f_zero();
    #pragma unroll
    for (int kk = 0; kk < 64; kk += 4) {
      v2f a; a.x = qs[ti * 16 + ml][kk + kb2]; a.y = qs[ti * 16 + ml][kk + kb2 + 1];
      v2f b; b.x = ks[tj * 16 + ml][kk + kb2]; b.y = ks[tj * 16 + ml][kk + kb2 + 1];
      acc = wmma_f32(a, b, acc);
    }
    #pragma unroll
    for (int r = 0; r < 8; r++) {
      int i = i0 + ti * 16 + r + hi * 8;
      int j = j0 + tj * 16 + ml;
      float sv = acc[r] * 0.125f;                 // 1/sqrt(64)
      S [base + (size_t)i * Tc + j] = sv;
      ST[base + (size_t)j * Tc + i] = sv;
    }
  }
}

// ---------------------------------------------------------------------------
// Kernel 3: per-key synaptic state -> a[j], m[j]. One wave per key j.
// grid (B*H*T/8), block 256.
// ---------------------------------------------------------------------------
__global__ __launch_bounds__(256)
void keystate_kernel(const float* __restrict__ S, const float* __restrict__ ST,
                     float* __restrict__ am) {
  const int gw = blockIdx.x * 8 + (threadIdx.x >> 5);
  const int lane = threadIdx.x & 31;
  const int bh = gw / Tc;
  const int j  = gw % Tc;
  const float* STr = ST + ((size_t)bh * Tc + j) * Tc;
  const float* Sr  = S  + ((size_t)bh * Tc + j) * Tc;

  // sweep 1: drive column sum  (softplus of clipped logits, i >= j)
  float ds = 0.f;
  for (int i = j + lane; i < Tc; i += 32) {
    float xv = fminf(fmaxf(STr[i], -20.f), 20.f);
    ds += (xv > 0.f) ? (xv + log1pf(expf(-xv))) : log1pf(expf(xv));
  }
  ds = warp_sum(ds);

  const float influx = ds / (float)(Tc - j);
  const float C   = 0.25f * influx;                       // C_new (>= 0)
  const float syt = 0.7f * C / (C + 0.4f) + 0.3f * C / (C + 3.0f);
  const float PRm = 0.6f * expf(-1.f / 18.f) + 0.04f;     // PR_mid (const)
  const float RRP = 0.8f * expf(-1.f / 40.f) + 0.008f;    // RRP_refill (const)
  const float g   = sigmoidf_(3.f * syt + 2.f * PRm - 2.f);
  const float a   = g * RRP;

  // sweep 2: raw release column total  (sum_i min(a * sigmoid(S[j,i]), 1))
  float rawsum = 0.f;
  for (int i = j + lane; i < Tc; i += 32) {
    rawsum += fminf(a * sigmoidf_(Sr[i]), 1.f);
  }
  rawsum = warp_sum(rawsum);

  const float total = fmaxf(rawsum, EPSf);
  const float scale = fminf(1.f, RRP / total);
  const float used  = rawsum * scale;
  const float RESn  = fminf(fmaxf(0.192f + used, 0.f), 1.f);
  const float Emid  = fminf(0.8f * expf(-1.f / 80.f) + 0.03f, 1.6f);
  const float En    = fminf(fmaxf(Emid - 0.015f * used - 0.006f * (1.f - RESn),
                                  0.f), 1.6f);
  const float qamp  = 1.3f * sigmoidf_(2.f * (En - 0.5f));
  const float m     = scale * fmaxf(qamp, EPSf);

  if (lane == 0) {
    am[(size_t)gw * 2 + 0] = a;
    am[(size_t)gw * 2 + 1] = m;
  }
}

// ---------------------------------------------------------------------------
// Kernel 4: causal softmax over (S + syn_logit); P overwrites S in place.
// One wave per query row. grid (B*H*T/8), block 256.
// ---------------------------------------------------------------------------
__global__ __launch_bounds__(256)
void softmax_kernel(const float* __restrict__ ST, const float* __restrict__ am,
                    float* __restrict__ S) {
  const int gw = blockIdx.x * 8 + (threadIdx.x >> 5);
  const int lane = threadIdx.x & 31;
  const int bh = gw / Tc;
  const int i  = gw % Tc;
  float*       Sr  = S  + ((size_t)bh * Tc + i) * Tc;
  const float* STr = ST + ((size_t)bh * Tc + i) * Tc;
  const float* amr = am + (size_t)bh * Tc * 2;
  const float distc = 0.075f / (float)Tc;

  // pass 1: online max/sum per lane
  float lmax = -1e30f, lsum = 0.f;
  for (int j = lane; j <= i; j += 32) {
    float aj = amr[2 * j], mj = amr[2 * j + 1];
    float rv = fminf(aj * sigmoidf_(STr[j]), 1.f);
    float l  = Sr[j] + logf(rv * mj + EPSf) - distc * (float)(i - j);
    if (l > lmax) { lsum = lsum * expf(lmax - l) + 1.f; lmax = l; }
    else          { lsum += expf(l - lmax); }
  }
  #pragma unroll
  for (int o = 16; o > 0; o >>= 1) {
    float m2 = __shfl_xor(lmax, o, 32);
    float s2 = __shfl_xor(lsum, o, 32);
    float M  = fmaxf(lmax, m2);
    lsum = lsum * expf(lmax - M) + s2 * expf(m2 - M);
    lmax = M;
  }
  const float inv = 1.f / lsum;

  // pass 2: write P, zero-padded to the 64-aligned K extent kernel 5 reads
  int jend = ((((i & ~15) + 16) + 63) >> 6) << 6;
  if (jend > Tc) jend = Tc;
  for (int j = lane; j < jend; j += 32) {
    float p = 0.f;
    if (j <= i) {
      float aj = amr[2 * j], mj = amr[2 * j + 1];
      float rv = fminf(aj * sigmoidf_(STr[j]), 1.f);
      float l  = Sr[j] + logf(rv * mj + EPSf) - distc * (float)(i - j);
      p = expf(l - lmax) * inv;
    }
    Sr[j] = p;
  }
}

// ---------------------------------------------------------------------------
// Kernel 5: ctx = P @ V (WMMA), output into (B,T,H*D) layout for Wo GEMM.
// grid (B*H, T/16), block 128 (4 waves, one 16-col slab each).
// ---------------------------------------------------------------------------
__global__ __launch_bounds__(128)
void ctx_kernel(const float* __restrict__ P, const float* __restrict__ v,
                float* __restrict__ ctx) {
  __shared__ float Ps[16][64];
  __shared__ float Vs[64][64];

  const int bh = blockIdx.x;
  const int i0 = blockIdx.y * 16;
  const int b = bh / Hc, h = bh % Hc;
  const int tid = threadIdx.x, lane = tid & 31, wave = tid >> 5;
  const int ml = lane & 15, kb2 = (lane >> 4) * 2, hi = lane >> 4;
  const int n0 = wave * 16;

  int klim = (((i0 + 16) + 63) >> 6) << 6;
  if (klim > Tc) klim = Tc;

  const float* Pb = P + (size_t)bh * Tc * Tc;
  const float* Vb = v + (size_t)bh * Tc * Dc;
  v8f acc = v8f_zero();

  for (int k0 = 0; k0 < klim; k0 += 64) {
    for (int e = tid; e < 256; e += 128) {    // P tile 16x64
      int r = e >> 4, c = (e & 15) * 4;
      cp_f4(&Ps[r][c], &Pb[(size_t)(i0 + r) * Tc + k0 + c]);
    }
    for (int e = tid; e < 1024; e += 128) {   // V tile 64x64
      int r = e >> 4, c = (e & 15) * 4;
      cp_f4(&Vs[r][c], &Vb[(size_t)(k0 + r) * Dc + c]);
    }
    cp_wait();
    __syncthreads();
    #pragma unroll
    for (int kk = 0; kk < 64; kk += 4) {
      v2f a; a.x = Ps[ml][kk + kb2];            a.y = Ps[ml][kk + kb2 + 1];
      v2f b; b.x = Vs[kk + kb2][n0 + ml];       b.y = Vs[kk + kb2 + 1][n0 + ml];
      acc = wmma_f32(a, b, acc);
    }
    __syncthreads();
  }

  #pragma unroll
  for (int r = 0; r < 8; r++) {
    int t = i0 + r + hi * 8;
    ctx[((size_t)b * Tc + t) * NEc + h * Dc + n0 + ml] = acc[r];
  }
}

// ---------------------------------------------------------------------------
// Kernel 6: y = ctx @ Wo. grid (B*T/16, NE/64), block 128 (4 waves).
// ---------------------------------------------------------------------------
__global__ __launch_bounds__(128)
void out_kernel(const float* __restrict__ ctx, const float* __restrict__ Wo,
                float* __restrict__ y) {
  __shared__ float As[16][16];
  __shared__ float Ws[16][64];

  const int row0 = blockIdx.x * 16;
  const int nbase = blockIdx.y * 64;
  const int tid = threadIdx.x, lane = tid & 31, wave = tid >> 5;
  const int ml = lane & 15, kb2 = (lane >> 4) * 2, hi = lane >> 4;
  const int n0 = wave * 16;

  v8f acc = v8f_zero();
  for (int k0 = 0; k0 < NEc; k0 += 16) {
    for (int e = tid; e < 64; e += 128) {     // ctx tile 16x16
      int r = e >> 2, c = (e & 3) * 4;
      cp_f4(&As[r][c], &ctx[(size_t)(row0 + r) * NEc + k0 + c]);
    }
    for (int e = tid; e < 256; e += 128) {    // Wo tile 16x64
      int r = e >> 4, c = (e & 15) * 4;
      cp_f4(&Ws[r][c], &Wo[(size_t)(k0 + r) * NEc + nbase + c]);
    }
    cp_wait();
    __syncthreads();
    #pragma unroll
    for (int kk = 0; kk < 16; kk += 4) {
      v2f a; a.x = As[ml][kk + kb2];        a.y = As[ml][kk + kb2 + 1];
      v2f b; b.x = Ws[kk + kb2][n0 + ml];   b.y = Ws[kk + kb2 + 1][n0 + ml];
      acc = wmma_f32(a, b, acc);
    }
    __syncthreads();
  }
  #pragma unroll
  for (int r = 0; r < 8; r++) {
    y[(size_t)(row0 + r + hi * 8) * NEc + nbase + n0 + ml] = acc[r];
  }
}

// ---------------------------------------------------------------------------
extern "C" void kernel_launch(void* const* d_in, const int* in_sizes, int n_in,
                              void* d_out, int out_size, void* d_ws,
                              size_t ws_size, hipStream_t stream) {
  const float* x  = (const float*)d_in[0];
  const float* Wq = (const float*)d_in[1];
  const float* Wk = (const float*)d_in[2];
  const float* Wv = (const float*)d_in[3];
  const float* Wo = (const float*)d_in[4];
  const float* rc = (const float*)d_in[5];
  const float* rs = (const float*)d_in[6];
  float* y = (float*)d_out;

  float* w = (float*)d_ws;
  const size_t nQKV = (size_t)Bc * Hc * Tc * Dc;   // 2,097,152
  const size_t nS   = (size_t)Bc * Hc * Tc * Tc;   // 33,554,432
  float* q   = w;                 w += nQKV;
  float* k   = w;                 w += nQKV;
  float* v   = w;                 w += nQKV;
  float* S   = w;                 w += nS;
  float* ST  = w;                 w += nS;
  float* am  = w;                 w += (size_t)Bc * Hc * Tc * 2;
  float* ctx = w;                 w += (size_t)Bc * Tc * NEc;

  qkv_kernel<<<dim3(Bc * Tc / 16, Hc), 128, 0, stream>>>(x, Wq, Wk, Wv, rc, rs,
                                                         q, k, v);
  scores_kernel<<<dim3(Bc * Hc, Tc / 64, Tc / 64), 256, 0, stream>>>(q, k, S,
                                                                     ST);
  keystate_kernel<<<dim3(Bc * Hc * Tc / 8), 256, 0, stream>>>(S, ST, am);
  softmax_kernel<<<dim3(Bc * Hc * Tc / 8), 256, 0, stream>>>(ST, am, S);
  ctx_kernel<<<dim3(Bc * Hc, Tc / 16), 128, 0, stream>>>(S, v, ctx);
  out_kernel<<<dim3(Bc * Tc / 16, NEc / 64), 128, 0, stream>>>(ctx, Wo, y);
}